// RandomOcclusions_7576322310611
// MI455X (gfx1250) — compile-verified
//
#include <hip/hip_runtime.h>

// RandomOcclusions: zero 16x16 patches of imgs[B,C,H,W] at (points_x, points_y).
// Single-pass, memory-bound design: per-image occlusion bitmask (H*W bits =
// 6.27KB) built in LDS via ds_or atomics, then one read + one write of the
// image stream with B128 vector ops. Traffic = 154MB -> ~6.6us @ 23.3 TB/s.

typedef float v4f __attribute__((ext_vector_type(4)));

constexpr int kB = 128;
constexpr int kC = 3;
constexpr int kH = 224;
constexpr int kW = 224;
constexpr int kP = 16;                     // patch size
constexpr int kHW = kH * kW;               // 50176 (bits in mask)
constexpr int kCHW = kC * kHW;             // 150528 floats per image
constexpr int kMaskWords = kHW / 32;       // 1568 u32 = 6272 B LDS
constexpr int kThreads = 256;              // 8 waves (wave32)
constexpr int kSplits = 49;                // chunks per image
constexpr int kF4PerBatch = kCHW / 4;      // 37632 float4 per image
constexpr int kF4PerBlock = kF4PerBatch / kSplits;   // 768
constexpr int kF4PerThread = kF4PerBlock / kThreads; // 3

__global__ __launch_bounds__(kThreads)
void occlude_kernel(const float* __restrict__ imgs,
                    const int* __restrict__ points_x,
                    const int* __restrict__ points_y,
                    float* __restrict__ out,
                    int n_points) {
    __shared__ unsigned int mask[kMaskWords];

    const int tid   = threadIdx.x;
    const int b     = blockIdx.y;   // image
    const int chunk = blockIdx.x;   // chunk within image

    // --- 1) clear mask (1568 words / 256 threads) ---
    for (int i = tid; i < kMaskWords; i += kThreads) mask[i] = 0u;
    __syncthreads();

    // --- 2) scatter patch rows into bitmask: 98 points x 16 rows ---
    const int* __restrict__ pxb = points_x + b * n_points;
    const int* __restrict__ pyb = points_y + b * n_points;
    const int tasks = n_points * kP;
    for (int t = tid; t < tasks; t += kThreads) {
        const int p = t >> 4;          // point index
        const int r = t & (kP - 1);    // row within patch
        const int x = pxb[p] + r;      // image row (H dim)
        const int y = pyb[p];          // start col (W dim)
        const int bit = x * kW + y;    // bit index in mask (row-major)
        const int w0  = bit >> 5;
        const int s   = bit & 31;
        atomicOr(&mask[w0], 0xFFFFu << s);                   // bits s..min(s+15,31)
        if (s > 16)
            atomicOr(&mask[w0 + 1], 0xFFFFu >> (32 - s));    // spill: low s-16 bits
    }
    __syncthreads();

    // --- 3) stream chunk: load B128 (NT), mask-select, store B128 ---
    const size_t base = (size_t)b * (size_t)kCHW;
    const v4f* __restrict__ in4 = (const v4f*)(imgs + base);
    v4f* __restrict__ out4      = (v4f*)(out + base);

    const int j0 = chunk * kF4PerBlock + tid;
#pragma unroll
    for (int k = 0; k < kF4PerThread; ++k) {
        const int j = j0 + k * kThreads;              // float4 index within image
        v4f v = __builtin_nontemporal_load(&in4[j]);  // read-once stream: TH=NT
        const int pix = (j * 4) % kHW;                // pixel of element 0 (4-aligned)
        const unsigned int bits = mask[pix >> 5] >> (pix & 31);
        if (bits & 0x1u) v.x = 0.0f;
        if (bits & 0x2u) v.y = 0.0f;
        if (bits & 0x4u) v.z = 0.0f;
        if (bits & 0x8u) v.w = 0.0f;
        out4[j] = v;
    }
}

extern "C" void kernel_launch(void* const* d_in, const int* in_sizes, int n_in,
                              void* d_out, int out_size, void* d_ws, size_t ws_size,
                              hipStream_t stream) {
    (void)n_in; (void)out_size; (void)d_ws; (void)ws_size;
    const float* imgs = (const float*)d_in[0];
    const int*   px   = (const int*)d_in[1];
    const int*   py   = (const int*)d_in[2];
    float*       out  = (float*)d_out;

    const int n_points = in_sizes[1] / kB;   // 98 for the reference shapes

    dim3 grid(kSplits, kB);
    dim3 block(kThreads);
    occlude_kernel<<<grid, block, 0, stream>>>(imgs, px, py, out, n_points);
}